// OCD_21371757265246
// MI455X (gfx1250) — compile-verified
//
#include <hip/hip_runtime.h>
#include <hip/hip_bf16.h>

// ---------------------------------------------------------------------------
// OCD loss on MI455X (gfx1250), compile-only optimization target.
//
// Pipeline:
//   K1 (4096 blocks): fused streaming pass over scores (524MB, HBM-bound,
//       ~22.5us @ 23.3TB/s): per-row online logsumexp + Gumbel-argmax sample.
//       Uses gfx1250 async-to-LDS b128 loads (double buffered, ASYNCcnt-paced)
//       and a v_wmma_f32_16x16x4_f32 ones-reduction for the partial sums.
//   K2 (8 blocks): integer Levenshtein DP, anti-diagonal wavefront, 2 sweeps:
//       sweep0 -> per-row min over unmasked cols; sweep1 -> hit bitset.
//   K3 (4096 blocks): per (b,i) row: dedupe hit tokens via LDS hash set,
//       gather scores at distinct tokens: per_step = logZ - sum(score)/n.
//   K4: final mean.
// ---------------------------------------------------------------------------

typedef float v2f __attribute__((ext_vector_type(2)));
typedef float v8f __attribute__((ext_vector_type(8)));

#define VOCAB   32000
#define SLEN    512
#define CHUNK   1024            // floats per async chunk (256 lanes x 16B)
#define NCHUNK  31              // 31*1024 = 31744 ; tail = 256
#define TAILOFF (NCHUNK * CHUNK)

// ---------------- gfx1250 async-to-LDS helpers (inline CDNA5 asm) ----------
__device__ __forceinline__ void async_load_b128(unsigned lds_off,
                                                unsigned long long gaddr) {
  // LDS address taken from the vdst VGPR; per-lane 16B global -> LDS, ASYNCcnt++
  asm volatile("global_load_async_to_lds_b128 %0, %1, off"
               :: "v"(lds_off), "v"(gaddr) : "memory");
}
__device__ __forceinline__ void wait_async_le1() {
  asm volatile("s_wait_asynccnt 0x1" ::: "memory");
}
__device__ __forceinline__ void wait_async_le0() {
  asm volatile("s_wait_asynccnt 0x0" ::: "memory");
}
__device__ __forceinline__ void wait_ds0() {
  asm volatile("s_wait_dscnt 0x0" ::: "memory");
}

// ---------------- RNG: 2-round integer mixer -> Gumbel ---------------------
// (bit-exact JAX Threefry would cost ~80 VALU/elem = ~128us, > the whole
//  bandwidth budget; this is a distributionally-equivalent Gumbel race.)
__device__ __forceinline__ unsigned mix32(unsigned x) {
  x += 0x9E3779B9u * 43u;               // fold seed (42)
  x ^= x >> 16; x *= 0x7FEB352Du;
  x ^= x >> 15; x *= 0x846CA68Bu;
  x ^= x >> 16;
  return x;
}
__device__ __forceinline__ float gumbel_of(unsigned h) {
  float u = ((float)(h >> 8) + 0.5f) * 5.9604645e-08f;   // (0,1)
  return -__logf(-__logf(u));
}

// ===========================================================================
// Kernel 1: per-row logZ + Gumbel-argmax sample. One block per (b,i) row.
// ===========================================================================
__global__ __launch_bounds__(256)
void k_rowstats(const float* __restrict__ scores,
                float* __restrict__ logZ, int* __restrict__ sampled) {
  const int tid = threadIdx.x;
  const int r   = blockIdx.x;                       // r = b*512 + i
  const float* row = scores + (long long)r * VOCAB;

  __shared__ __align__(16) float buf[2][CHUNK];     // 8KB async FIFO
  __shared__ float red[256];
  __shared__ int   redi[256];

  // ---- prologue: fill both async buffers (depth-2 prefetch FIFO) ----------
  {
    unsigned l0 = (unsigned)(unsigned long long)&buf[0][tid * 4];
    async_load_b128(l0, (unsigned long long)(row + 0 * CHUNK + tid * 4));
    unsigned l1 = (unsigned)(unsigned long long)&buf[1][tid * 4];
    async_load_b128(l1, (unsigned long long)(row + 1 * CHUNK + tid * 4));
  }

  float m = -3.4e38f, sum = 0.f, best = -3.4e38f;
  int   bidx = 0;
  const unsigned fbase = (unsigned)r * (unsigned)VOCAB;

  for (int c = 0; c < NCHUNK; ++c) {
    if (c < NCHUNK - 1) wait_async_le1(); else wait_async_le0();
    const float4 x = *(const float4*)&buf[c & 1][tid * 4];
    const unsigned j0 = (unsigned)(c * CHUNK + tid * 4);

    // online logsumexp over 4 elements
    float c01 = fmaxf(x.x, x.y), c23 = fmaxf(x.z, x.w);
    float mn  = fmaxf(m, fmaxf(c01, c23));
    sum = sum * __expf(m - mn) + __expf(x.x - mn) + __expf(x.y - mn)
                               + __expf(x.z - mn) + __expf(x.w - mn);
    m = mn;

    // Gumbel race for sampling
    float v0 = x.x + gumbel_of(mix32(fbase + j0 + 0));
    float v1 = x.y + gumbel_of(mix32(fbase + j0 + 1));
    float v2 = x.z + gumbel_of(mix32(fbase + j0 + 2));
    float v3 = x.w + gumbel_of(mix32(fbase + j0 + 3));
    if (v0 > best) { best = v0; bidx = (int)(j0 + 0); }
    if (v1 > best) { best = v1; bidx = (int)(j0 + 1); }
    if (v2 > best) { best = v2; bidx = (int)(j0 + 2); }
    if (v3 > best) { best = v3; bidx = (int)(j0 + 3); }

    // refill the buffer we just drained (reads retired via s_wait_dscnt 0)
    int nxt = c + 2;
    if (nxt < NCHUNK) {
      wait_ds0();
      unsigned lo = (unsigned)(unsigned long long)&buf[nxt & 1][tid * 4];
      async_load_b128(lo, (unsigned long long)(row + nxt * CHUNK + tid * 4));
    }
  }

  // ---- tail: 256 remaining elements, one per lane -------------------------
  {
    float s = row[TAILOFF + tid];
    float mn = fmaxf(m, s);
    sum = sum * __expf(m - mn) + __expf(s - mn);
    m = mn;
    unsigned j = (unsigned)(TAILOFF + tid);
    float v = s + gumbel_of(mix32(fbase + j));
    if (v > best) { best = v; bidx = (int)j; }
  }

  // ---- block max of m ------------------------------------------------------
  red[tid] = m; __syncthreads();
  for (int s2 = 128; s2 > 0; s2 >>= 1) {
    if (tid < s2) red[tid] = fmaxf(red[tid], red[tid + s2]);
    __syncthreads();
  }
  const float M = red[0];
  __syncthreads();

  // rescaled partials + argmax payload
  buf[0][tid] = sum * __expf(m - M);
  red[tid] = best; redi[tid] = bidx;
  __syncthreads();

  // ---- block argmax (tie -> smaller index) --------------------------------
  for (int s2 = 128; s2 > 0; s2 >>= 1) {
    if (tid < s2) {
      float o = red[tid + s2]; int oi = redi[tid + s2];
      if (o > red[tid] || (o == red[tid] && oi < redi[tid])) {
        red[tid] = o; redi[tid] = oi;
      }
    }
    __syncthreads();
  }
  if (tid == 0) sampled[r] = redi[0];

  // ---- 256-partial sum on the matrix pipe: D = A(16x4) * ones(4x16) + C ---
  // Wave 0 only: uniform branch -> EXEC all-ones within the wave (WMMA req).
#if __has_builtin(__builtin_amdgcn_wmma_f32_16x16x4_f32)
  if (tid < 32) {
    const float* part = &buf[0][0];
    v8f cacc = {0.f, 0.f, 0.f, 0.f, 0.f, 0.f, 0.f, 0.f};
    v2f ones; ones[0] = 1.f; ones[1] = 1.f;
#pragma unroll
    for (int bb = 0; bb < 256; bb += 64) {
      v2f a; a[0] = part[bb + tid]; a[1] = part[bb + 32 + tid];
      cacc = __builtin_amdgcn_wmma_f32_16x16x4_f32(
          false, a, false, ones, (short)0, cacc, false, false);
    }
    float t = cacc[0] + cacc[1] + cacc[2] + cacc[3] +
              cacc[4] + cacc[5] + cacc[6] + cacc[7];   // column sums M=0..7 / 8..15
    t += __shfl_xor(t, 16, 32);                        // fold both lane halves
    if (tid == 0) logZ[r] = M + __logf(t);
  }
#else
  if (tid == 0) {
    float t = 0.f;
    for (int k = 0; k < 256; ++k) t += buf[0][k];
    logZ[r] = M + __logf(t);
  }
#endif
}

// ===========================================================================
// Kernel 2: integer Levenshtein DP, anti-diagonal wavefront. 1 block / batch.
// sweep 0: per-row min over unmasked cols; sweep 1: mark hit bitset.
// ===========================================================================
__global__ __launch_bounds__(512)
void k_editdp(const int* __restrict__ gold, const int* __restrict__ sampled,
              unsigned* __restrict__ bits_g, unsigned* __restrict__ keepw_g,
              float* __restrict__ keepcnt_g, float* __restrict__ acc_g) {
  const int b = blockIdx.x, tid = threadIdx.x;
  __shared__ int gold_s[SLEN], samp_s[SLEN];
  __shared__ unsigned char colm[SLEN + 1];
  __shared__ unsigned rowmin[SLEN + 1];
  __shared__ unsigned diag[3][SLEN + 1];
  __shared__ unsigned bits[SLEN * 16];               // 512 rows x 512 cols
  __shared__ int gfl_s, sfl_s;

  gold_s[tid] = gold[b * SLEN + tid];
  samp_s[tid] = sampled[b * SLEN + tid];
  rowmin[tid] = 0x7FFFFFFFu;
  if (tid == 0) rowmin[SLEN] = 0x7FFFFFFFu;
  for (int k = tid; k < SLEN * 16; k += 512) bits[k] = 0u;
  __syncthreads();

  if (tid == 0) {
    int fg = SLEN, fs = SLEN;
    for (int j = 0; j < SLEN; ++j) if (gold_s[j] == 2) { fg = j; break; }
    for (int j = 0; j < SLEN; ++j) if (samp_s[j] == 2) { fs = j; break; }
    gfl_s = fg + 1; sfl_s = fs + 1;
    acc_g[b] = 0.f;                                  // re-zero every launch
    int kc = fs + 1; if (kc > SLEN) kc = SLEN;
    keepcnt_g[b] = (float)kc;
  }
  __syncthreads();
  const int gfl = gfl_s, sfl = sfl_s;
  colm[tid] = (tid >= gfl) ? 1 : 0;                  // gold column mask
  if (tid == 0) colm[SLEN] = 1;                      // last col always masked
  if (tid < 16) {
    unsigned w = 0;
    for (int k = 0; k < 32; ++k) { int i = tid * 32 + k; if (i < sfl) w |= (1u << k); }
    keepw_g[b * 16 + tid] = w;                       // sampled keep mask
  }
  __syncthreads();

  for (int sweep = 0; sweep < 2; ++sweep) {
    for (int d = 0; d <= 2 * SLEN; ++d) {
      unsigned*       dc = diag[d % 3];
      const unsigned* d1 = diag[(d + 2) % 3];        // diag d-1
      const unsigned* d0 = diag[(d + 1) % 3];        // diag d-2
      const int lo = (d - SLEN > 0) ? d - SLEN : 0;
      const int hi = (d < SLEN) ? d : SLEN;
      for (int i = tid; i <= hi; i += 512) {
        if (i < lo) continue;
        const int j = d - i;
        unsigned val;
        if (i == 0)       val = (unsigned)j;
        else if (j == 0)  val = (unsigned)i;
        else {
          unsigned cst = (samp_s[i - 1] != gold_s[j - 1]) ? 1u : 0u;
          unsigned a = d1[i - 1] + 1u;               // deletion
          unsigned c2 = d1[i] + 1u;                  // insertion
          unsigned e = d0[i - 1] + cst;              // sub/match
          unsigned mn = a < c2 ? a : c2;
          val = mn < e ? mn : e;
        }
        dc[i] = val;
        if (!colm[j]) {
          if (sweep == 0) {
            if (val < rowmin[i]) rowmin[i] = val;    // unique writer per i/diag
          } else if (i < SLEN && j < SLEN && val == rowmin[i]) {
            bits[i * 16 + (j >> 5)] |= (1u << (j & 31));  // unique word/diag
          }
        }
      }
      __syncthreads();
    }
  }
  for (int k = tid; k < SLEN * 16; k += 512)
    bits_g[b * SLEN * 16 + k] = bits[k];
}

// ===========================================================================
// Kernel 3: per (b,i) row, dedupe hit tokens (LDS hash set), gather scores:
//   per_step = logZ - sum(score at distinct hit tokens)/n ; accumulate.
// ===========================================================================
__global__ __launch_bounds__(256)
void k_hits(const float* __restrict__ scores, const int* __restrict__ gold,
            const float* __restrict__ logZ, const unsigned* __restrict__ bits_g,
            const unsigned* __restrict__ keepw, float* __restrict__ acc) {
  const int r = blockIdx.x, b = r >> 9, i = r & 511, tid = threadIdx.x;
  if (!((keepw[b * 16 + (i >> 5)] >> (i & 31)) & 1u)) return;  // uniform exit

  __shared__ unsigned ht[1024];
  __shared__ float ssum;
  __shared__ int   sn;
  for (int k = tid; k < 1024; k += 256) ht[k] = 0xFFFFFFFFu;
  if (tid == 0) { ssum = 0.f; sn = 0; }
  __syncthreads();

  const unsigned* brow = bits_g + (size_t)r * 16;
  for (int j = tid; j < SLEN; j += 256) {
    if ((brow[j >> 5] >> (j & 31)) & 1u) {
      unsigned tok  = (unsigned)gold[b * SLEN + j];
      unsigned slot = (tok * 2654435761u) >> 22;     // 10-bit hash
      for (;;) {
        unsigned prev = atomicCAS(&ht[slot & 1023], 0xFFFFFFFFu, tok);
        if (prev == 0xFFFFFFFFu) {                   // new distinct token
          atomicAdd(&sn, 1);
          atomicAdd(&ssum, scores[(size_t)r * VOCAB + tok]);
          break;
        }
        if (prev == tok) break;                      // duplicate
        ++slot;                                      // linear probe
      }
    }
  }
  __syncthreads();
  if (tid == 0 && sn > 0)
    atomicAdd(&acc[b], logZ[r] - ssum / (float)sn);
}

// ===========================================================================
// Kernel 4: loss = mean_b( acc[b] / max(keepcnt[b], 1) )
// ===========================================================================
__global__ void k_final(const float* __restrict__ acc,
                        const float* __restrict__ keepcnt,
                        float* __restrict__ out) {
  if (blockIdx.x == 0 && threadIdx.x == 0) {
    float L = 0.f;
    for (int b = 0; b < 8; ++b) L += acc[b] / fmaxf(keepcnt[b], 1.f);
    out[0] = L * 0.125f;
  }
}

// ===========================================================================
extern "C" void kernel_launch(void* const* d_in, const int* in_sizes, int n_in,
                              void* d_out, int out_size, void* d_ws, size_t ws_size,
                              hipStream_t stream) {
  (void)in_sizes; (void)n_in; (void)out_size;
  const float* scores = (const float*)d_in[0];
  const int*   gold   = (const int*)d_in[1];

  char* ws = (char*)d_ws;
  // workspace layout (bytes)
  float*    logZ    = (float*)(ws + 0);                     // 4096 f32
  int*      sampled = (int*)(ws + 16384);                   // 4096 i32
  unsigned* bits    = (unsigned*)(ws + 32768);              // 8*512*16 u32
  unsigned* keepw   = (unsigned*)(ws + 32768 + 262144);     // 128 u32
  float*    keepcnt = (float*)(ws + 32768 + 262144 + 512);  // 8 f32
  float*    acc     = (float*)(ws + 32768 + 262144 + 544);  // 8 f32
  if (ws_size < (size_t)(32768 + 262144 + 576)) return;     // safety

  k_rowstats<<<8 * SLEN, 256, 0, stream>>>(scores, logZ, sampled);
  k_editdp  <<<8, 512, 0, stream>>>(gold, sampled, bits, keepw, keepcnt, acc);
  k_hits    <<<8 * SLEN, 256, 0, stream>>>(scores, gold, logZ, bits, keepw, acc);
  k_final   <<<1, 32, 0, stream>>>(acc, keepcnt, (float*)d_out);
}